// CUDALIFNeuron_1365799600601
// MI455X (gfx1250) — compile-verified
//
#include <hip/hip_runtime.h>
#include <stdint.h>

// LIF scan over T timesteps of N neurons.
//   mem = BETA*mem + x[t]; spk = (mem >= THR); mem -= THR*spk; out[t] = spk
// Memory-bound: ~420 MB moved -> ~18us floor at 23.3 TB/s on MI455X.
// Data path: CDNA5 async global->LDS copy engine (ASYNCcnt), 8-deep LDS
// pipeline per wave (4 KB in flight/wave), distance-16 L2 prefetch
// (global_prefetch_b8), ds_load_b128 consume, non-temporal b128 stores
// (420 MB stream > 192 MB L2).

#define LIF_T 100
#define LIF_N (32 * 16384)   // neurons per timestep
#define TPB   256            // 8 waves (wave32) per block
#define VEC   4              // floats per thread per timestep (b128)
#define PD    8              // async pipeline depth (LDS slots), power of 2
#define PFD   16             // L2 prefetch distance (timesteps)

typedef __attribute__((ext_vector_type(4))) float v4f;

__device__ __forceinline__ void lif_step(v4f& mem, const v4f xt, v4f& spk) {
    mem.x = 0.25f * mem.x + xt.x; spk.x = (mem.x >= 1.0f) ? 1.0f : 0.0f; mem.x -= spk.x;
    mem.y = 0.25f * mem.y + xt.y; spk.y = (mem.y >= 1.0f) ? 1.0f : 0.0f; mem.y -= spk.y;
    mem.z = 0.25f * mem.z + xt.z; spk.z = (mem.z >= 1.0f) ? 1.0f : 0.0f; mem.z -= spk.z;
    mem.w = 0.25f * mem.w + xt.w; spk.w = (mem.w >= 1.0f) ? 1.0f : 0.0f; mem.w -= spk.w;
}

__global__ __launch_bounds__(TPB) void lif_scan_kernel(const float* __restrict__ x,
                                                       float* __restrict__ out) {
    __shared__ v4f buf[PD][TPB];   // 32 KB of the 320 KB/WGP LDS

    const int tid = threadIdx.x;
    const long long n0 = (long long)blockIdx.x * (TPB * VEC) + (long long)tid * VEC;
    const v4f* __restrict__ xp = reinterpret_cast<const v4f*>(x + n0);
    v4f* __restrict__ op       = reinterpret_cast<v4f*>(out + n0);
    const int stride4 = LIF_N / VEC;   // v4f elements between timesteps

    // Wave-relative LDS byte addresses for this lane's PD slots.
    // Flat LDS aperture base occupies addr[63:32]; low 32 bits == LDS offset.
    uint32_t ldsOff[PD];
#pragma unroll
    for (int i = 0; i < PD; ++i)
        ldsOff[i] = (uint32_t)(uintptr_t)(&buf[i][tid]);

    // ---- Prologue: fill the pipeline with PD async copies (ASYNCcnt). ----
#pragma unroll
    for (int i = 0; i < PD; ++i) {
        asm volatile("global_load_async_to_lds_b128 %0, %1, off"
                     :: "v"(ldsOff[i]), "v"(xp + (long long)i * stride4)
                     : "memory");
    }

    v4f mem = {0.f, 0.f, 0.f, 0.f};

    // ---- Steady state: branch-free body, always PD loads in flight. ----
    for (int t = 0; t < LIF_T - PD; ++t) {
        // Async loads complete in order: <= PD-1 outstanding => load t landed.
        asm volatile("s_wait_asynccnt 7" ::: "memory");

        v4f xt = buf[t & (PD - 1)][tid];          // ds_load_b128

        // DS read must retire before the async engine may overwrite the slot
        // (async LDS writes are unordered vs the DS pipe).
        asm volatile("s_wait_dscnt 0" ::: "memory");

        // Refill this slot with timestep t+PD (always valid: t+PD < LIF_T).
        asm volatile("global_load_async_to_lds_b128 %0, %1, off"
                     :: "v"(ldsOff[t & (PD - 1)]),
                        "v"(xp + (long long)(t + PD) * stride4)
                     : "memory");

        // Warm L2 beyond the async pipeline depth; branchless clamp keeps the
        // address in-bounds (re-prefetching the last step is harmless).
        int pf = t + PFD;
        pf = (pf < LIF_T - 1) ? pf : (LIF_T - 1);
        __builtin_prefetch((const void*)(xp + (long long)pf * stride4), 0, 0);

        v4f spk;
        lif_step(mem, xt, spk);

        // Streaming output: non-temporal b128 store (don't pollute L2).
        __builtin_nontemporal_store(spk, op + (long long)t * stride4);
    }

    // ---- Epilogue: drain the pipeline once, then consume the last PD. ----
    asm volatile("s_wait_asynccnt 0" ::: "memory");
#pragma unroll
    for (int e = 0; e < PD; ++e) {
        const int t = LIF_T - PD + e;
        v4f xt = buf[t & (PD - 1)][tid];
        v4f spk;
        lif_step(mem, xt, spk);
        __builtin_nontemporal_store(spk, op + (long long)t * stride4);
    }
}

extern "C" void kernel_launch(void* const* d_in, const int* in_sizes, int n_in,
                              void* d_out, int out_size, void* d_ws, size_t ws_size,
                              hipStream_t stream) {
    (void)in_sizes; (void)n_in; (void)d_ws; (void)ws_size; (void)out_size;
    const float* x = (const float*)d_in[0];
    float* out = (float*)d_out;
    const int blocks = LIF_N / (TPB * VEC);   // 512
    lif_scan_kernel<<<blocks, TPB, 0, stream>>>(x, out);
}